// MultiVarModel_48000554500814
// MI455X (gfx1250) — compile-verified
//
#include <hip/hip_runtime.h>

// Celerite O(N) log-likelihood as two hierarchical projective (Mobius) scans.
//   cS recursion:  cS' = (m00*cS + m01)/(m10*cS + m11)  with
//                  M_n = [[(a-2UV)P^2, V^2], [-U^2 P^2, a]]
//   cg recursion:  linear  -> matrices [[alpha,beta],[0,1]], same machinery.
// Block-level combine of 2x2 matrices is done with V_WMMA_F32_16X16X4_F32:
// 8 independent 2x2 products packed block-diagonally into one 16x16x16 matmul
// (4 chained K=4 WMMAs, exact f32 accumulation).

#define EPT   8
#define BLOCK 256
#define CHUNK (EPT * BLOCK)   // 2048 elements per block

typedef __attribute__((ext_vector_type(2))) float v2f;
typedef __attribute__((ext_vector_type(8))) float v8f;

// v_rcp_f32 + one Newton step: ~0.5 ulp, much shorter than the IEEE div expansion.
__device__ __forceinline__ float fast_rcp(float d) {
  float r = __builtin_amdgcn_rcpf(d);
  float e = __builtin_fmaf(-d, r, 1.0f);
  return __builtin_fmaf(e, r, r);
}

__device__ __forceinline__ float mob_apply(const float* m, float s) {
  return (m[0] * s + m[1]) * fast_rcp(m[2] * s + m[3]);
}

// m <- E * m (E applied after m). No normalization (bounded growth over EPT=8).
__device__ __forceinline__ void mat_mul(float* m, float e00, float e01,
                                        float e10, float e11) {
  float c00 = e00 * m[0] + e01 * m[2];
  float c01 = e00 * m[1] + e01 * m[3];
  float c10 = e10 * m[0] + e11 * m[2];
  float c11 = e10 * m[1] + e11 * m[3];
  m[0] = c00; m[1] = c01; m[2] = c10; m[3] = c11;
}

// Projective max-abs normalization; raw v_rcp (precision irrelevant here).
__device__ __forceinline__ void mat_norm(float* m) {
  float nrm = fmaxf(fmaxf(fabsf(m[0]), fabsf(m[1])),
                    fmaxf(fabsf(m[2]), fabsf(m[3])));
  float inv = __builtin_amdgcn_rcpf(fmaxf(nrm, 1e-30f));
  m[0] *= inv; m[1] *= inv; m[2] *= inv; m[3] *= inv;
}

// One wave combines up to 8 adjacent pairs of 2x2 matrices from LDS:
//   dst[j] = src[2j+1] * src[2j]   (later-in-time on the left)
// via a block-diagonal 16x16 x 16x16 product = 4x V_WMMA_F32_16X16X4_F32.
// Fragment layouts per CDNA5 ISA 7.12.2 (f32 A 16x4, B 4x16, C/D 16x16).
// Loads are unconditional (indices clamped to valid matrices); ownership is
// applied as a value mask so no exec-predicated ds_loads are generated.
__device__ __forceinline__ void wmma_pair_combine(const float* src, float* dst,
                                                  int cnt) {
  int lane = (int)(threadIdx.x & 31);
  int p  = lane & 15;    // row (A), col (B/D)
  int h  = lane >> 4;    // half select
  int pr = p & 1;        // local row/col within a 2x2 block
  int pb = p >> 1;       // block index owned by this lane's row/col
  v8f acc = {};
#pragma unroll
  for (int kc = 0; kc < 4; ++kc) {
    int bcol = 2 * kc + h;                 // diagonal block touched by this K-slice
    int bi   = (bcol < cnt) ? bcol : 0;    // clamp: loads always hit valid data
    const float* L = src + (2 * bi + 1) * 4;
    const float* R = src + (2 * bi) * 4;
    float ownf = ((pb == bcol) && (bcol < cnt)) ? 1.0f : 0.0f;
    v2f af, bf;
    // A[p][4kc+2h+v] = L_bcol[pr][v]  (contiguous pair -> ds_load_b64)
    af.x = L[pr * 2 + 0] * ownf;
    af.y = L[pr * 2 + 1] * ownf;
    // B[4kc+2h+r][p] = R_bcol[r][pr]
    bf.x = R[0 + pr] * ownf;
    bf.y = R[2 + pr] * ownf;
    acc = __builtin_amdgcn_wmma_f32_16x16x4_f32(false, af, false, bf,
                                                (short)0, acc, false, false);
  }
  // D block j lives at rows 2j,2j+1 (vgpr 2(j&3), half j>>2), cols 2j,2j+1.
  int j = pb;
  if ((h == (j >> 2)) && (j < cnt)) {
    float a0 = (j & 1) ? acc[2] : acc[0];
    float a1 = (j & 1) ? acc[3] : acc[1];
    float b0 = (j & 1) ? acc[6] : acc[4];
    float b1 = (j & 1) ? acc[7] : acc[5];
    float e0 = (j & 2) ? b0 : a0;   // D[2j][p]
    float e1 = (j & 2) ? b1 : a1;   // D[2j+1][p]
    float mx = fmaxf(fabsf(e0), fabsf(e1));
    float om = __shfl_xor(mx, 1, 32);          // partner lane = other column
    float inv = __builtin_amdgcn_rcpf(fmaxf(fmaxf(mx, om), 1e-30f));
    float* o = dst + j * 4 + pr;               // col0 lane: m00,m10; col1: m01,m11
    o[0] = e0 * inv;
    o[2] = e1 * inv;
  }
}

// Tree-reduce 256 per-thread matrices (order-preserving) to one block matrix.
__device__ void mat_tree(float* bufA, float* bufB, float* out4) {
  float* cur = bufA;
  float* nxt = bufB;
  int n = BLOCK;
  while (n > 1) {
    int nprod = n >> 1;
    int wave = (int)(threadIdx.x >> 5);
    int ngroups = (nprod + 7) >> 3;
    for (int g = wave; g < ngroups; g += (BLOCK / 32)) {
      int j0 = g << 3;
      int cnt = nprod - j0; if (cnt > 8) cnt = 8;
      wmma_pair_combine(cur + j0 * 8, nxt + j0 * 4, cnt);
    }
    __syncthreads();
    float* t = cur; cur = nxt; nxt = t;
    n = nprod;
  }
  if (threadIdx.x == 0) {
    out4[0] = cur[0]; out4[1] = cur[1]; out4[2] = cur[2]; out4[3] = cur[3];
  }
}

__device__ __forceinline__ void load_params(const float* lad, const float* lkp,
                                            float* amps, float& sigma2,
                                            float& invell) {
  amps[0] = 1.0f;
  amps[1] = __expf(lad[0]);
  amps[2] = __expf(lad[1]);
  amps[3] = __expf(lad[2]);
  sigma2 = __expf(2.0f * lkp[0]);
  invell = __expf(-lkp[1]);
}

__device__ __forceinline__ void stage_chunk(const float* t, const int* band,
                                            const float* yerr, const float* y,
                                            float* sP, float* sA, float* sV,
                                            float* sY, const float* amps,
                                            float sigma2, float invell,
                                            int base, int N) {
  for (int i = (int)threadIdx.x; i < CHUNK; i += BLOCK) {
    int gi = base + i;
    if (gi < N) {
      float tt = t[gi];
      float tp = (gi > 0) ? t[gi - 1] : tt;
      float u  = amps[band[gi] & 3];
      float ye = yerr[gi];
      sP[i] = (gi > 0) ? __expf(-(tt - tp) * invell) : 0.0f;
      sA[i] = ye * ye + sigma2 * u * u;
      sV[i] = u;
      if (sY) sY[i] = y[gi];
    } else {
      sP[i] = 0.0f; sA[i] = 1.0f; sV[i] = 0.0f;
      if (sY) sY[i] = 0.0f;
    }
  }
}

// Compose the EPT element Mobius matrices of this thread; normalize once.
template <bool GUARD>
__device__ __forceinline__ void thread_mobius(const float* sP, const float* sA,
                                              const float* sV, float sigma2,
                                              int base, int N, float* mats) {
  float m[4] = {1.f, 0.f, 0.f, 1.f};
  int tb = (int)threadIdx.x * EPT;
#pragma unroll
  for (int e = 0; e < EPT; ++e) {
    int i = tb + e;
    if (!GUARD || (base + i < N)) {
      float P = sP[i], a = sA[i], V = sV[i], U = sigma2 * V;
      float P2 = P * P;
      mat_mul(m, (a - 2.0f * U * V) * P2, V * V, -U * U * P2, a);
    }
  }
  mat_norm(m);
  mats[threadIdx.x * 4 + 0] = m[0];
  mats[threadIdx.x * 4 + 1] = m[1];
  mats[threadIdx.x * 4 + 2] = m[2];
  mats[threadIdx.x * 4 + 3] = m[3];
}

// Walk with known S prefix: emit per-thread linear matrix, return logD partial.
template <bool GUARD>
__device__ __forceinline__ float walk_linear(const float* sP, const float* sA,
                                             const float* sV, const float* sY,
                                             float sigma2, float cS, int base,
                                             int N, float* lout) {
  float l[4] = {1.f, 0.f, 0.f, 1.f};
  float logDs = 0.0f;
  int tb = (int)threadIdx.x * EPT;
#pragma unroll
  for (int e = 0; e < EPT; ++e) {
    int i = tb + e;
    if (!GUARD || (base + i < N)) {
      float P = sP[i], a = sA[i], V = sV[i], U = sigma2 * V;
      float Sn = P * P * cS;
      float D = a - U * U * Sn;
      float invD = fast_rcp(D);
      float W = (V - Sn * U) * invD;
      logDs += __logf(D);
      float al = P * (1.0f - W * U);
      float be = W * sY[i];
      mat_mul(l, al, be, 0.0f, 1.0f);
      cS = Sn + D * W * W;
    }
  }
  mat_norm(l);
  lout[0] = l[0]; lout[1] = l[1]; lout[2] = l[2]; lout[3] = l[3];
  return logDs;
}

// Final walk with known S and g prefixes: return quad partial.
template <bool GUARD>
__device__ __forceinline__ float walk_quad(const float* sP, const float* sA,
                                           const float* sV, const float* sY,
                                           float sigma2, float cS, float cg,
                                           int base, int N) {
  float quad = 0.0f;
  int tb = (int)threadIdx.x * EPT;
#pragma unroll
  for (int e = 0; e < EPT; ++e) {
    int i = tb + e;
    if (!GUARD || (base + i < N)) {
      float P = sP[i], a = sA[i], V = sV[i], U = sigma2 * V;
      float Sn = P * P * cS;
      float D = a - U * U * Sn;
      float invD = fast_rcp(D);
      float W = (V - Sn * U) * invD;
      float gn = P * cg;
      float z = sY[i] - U * gn;
      quad += z * z * invD;
      cS = Sn + D * W * W;
      cg = gn + W * z;
    }
  }
  return quad;
}

// ---- Kernel A: per-block composed Mobius matrix ----------------------------
__global__ void __launch_bounds__(BLOCK) k_blockA(
    const float* __restrict__ t, const int* __restrict__ band,
    const float* __restrict__ yerr, const float* __restrict__ lad,
    const float* __restrict__ lkp, float* __restrict__ blockMob, int N) {
  __shared__ float sP[CHUNK], sA[CHUNK], sV[CHUNK];
  __shared__ float mA[BLOCK * 4], mB[BLOCK * 4];
  float amps[4], sigma2, invell;
  load_params(lad, lkp, amps, sigma2, invell);
  int base = (int)blockIdx.x * CHUNK;
  stage_chunk(t, band, yerr, nullptr, sP, sA, sV, nullptr, amps, sigma2, invell, base, N);
  __syncthreads();
  if (base + CHUNK <= N) thread_mobius<false>(sP, sA, sV, sigma2, base, N, mA);
  else                   thread_mobius<true>(sP, sA, sV, sigma2, base, N, mA);
  __syncthreads();
  mat_tree(mA, mB, blockMob + blockIdx.x * 4);
}

// ---- Kernel B: global exclusive combine of S (sequential over ~1k blocks) --
__global__ void k_combS(const float* __restrict__ blockMob,
                        float* __restrict__ blockS0, int nb) {
  if (threadIdx.x == 0 && blockIdx.x == 0) {
    float s = 0.0f;
    for (int k = 0; k < nb; ++k) {
      blockS0[k] = s;
      s = mob_apply(blockMob + k * 4, s);
    }
  }
}

// ---- Kernel C: per-block linear (alpha,beta) summary + logD partials -------
__global__ void __launch_bounds__(BLOCK) k_blockC(
    const float* __restrict__ t, const int* __restrict__ band,
    const float* __restrict__ y, const float* __restrict__ yerr,
    const float* __restrict__ lad, const float* __restrict__ lkp,
    const float* __restrict__ blockS0, float* __restrict__ blockLin,
    float* __restrict__ blockLogD, int N) {
  __shared__ float sP[CHUNK], sA[CHUNK], sV[CHUNK], sY[CHUNK];
  __shared__ float mA[BLOCK * 4], mB[BLOCK * 4];
  __shared__ float sS0[BLOCK];
  __shared__ float red[BLOCK];
  float amps[4], sigma2, invell;
  load_params(lad, lkp, amps, sigma2, invell);
  int base = (int)blockIdx.x * CHUNK;
  int tid = (int)threadIdx.x;
  bool full = (base + CHUNK <= N);
  stage_chunk(t, band, yerr, y, sP, sA, sV, sY, amps, sigma2, invell, base, N);
  __syncthreads();
  if (full) thread_mobius<false>(sP, sA, sV, sigma2, base, N, mA);
  else      thread_mobius<true>(sP, sA, sV, sigma2, base, N, mA);
  __syncthreads();
  if (tid == 0) {   // exclusive S-prefix at each thread's sub-chunk start
    float s = blockS0[blockIdx.x];
    for (int i = 0; i < BLOCK; ++i) { sS0[i] = s; s = mob_apply(mA + i * 4, s); }
  }
  __syncthreads();
  float cS = sS0[tid];
  float logDs;
  {
    float l[4];
    if (full) logDs = walk_linear<false>(sP, sA, sV, sY, sigma2, cS, base, N, l);
    else      logDs = walk_linear<true>(sP, sA, sV, sY, sigma2, cS, base, N, l);
    mA[tid * 4 + 0] = l[0]; mA[tid * 4 + 1] = l[1];
    mA[tid * 4 + 2] = l[2]; mA[tid * 4 + 3] = l[3];
  }
  __syncthreads();
  mat_tree(mA, mB, blockLin + blockIdx.x * 4);
  red[tid] = logDs;
  __syncthreads();
  for (int s = BLOCK / 2; s > 0; s >>= 1) {
    if (tid < s) red[tid] += red[tid + s];
    __syncthreads();
  }
  if (tid == 0) blockLogD[blockIdx.x] = red[0];
}

// ---- Kernel D: global exclusive combine of g -------------------------------
__global__ void k_combG(const float* __restrict__ blockLin,
                        float* __restrict__ blockG0, int nb) {
  if (threadIdx.x == 0 && blockIdx.x == 0) {
    float g = 0.0f;
    for (int k = 0; k < nb; ++k) {
      blockG0[k] = g;
      g = mob_apply(blockLin + k * 4, g);
    }
  }
}

// ---- Kernel E: final replay -> quad partials -------------------------------
__global__ void __launch_bounds__(BLOCK) k_blockE(
    const float* __restrict__ t, const int* __restrict__ band,
    const float* __restrict__ y, const float* __restrict__ yerr,
    const float* __restrict__ lad, const float* __restrict__ lkp,
    const float* __restrict__ blockS0, const float* __restrict__ blockG0,
    float* __restrict__ blockQuad, int N) {
  __shared__ float sP[CHUNK], sA[CHUNK], sV[CHUNK], sY[CHUNK];
  __shared__ float mA[BLOCK * 4];
  __shared__ float sS0[BLOCK], sG0[BLOCK];
  __shared__ float red[BLOCK];
  float amps[4], sigma2, invell;
  load_params(lad, lkp, amps, sigma2, invell);
  int base = (int)blockIdx.x * CHUNK;
  int tid = (int)threadIdx.x;
  bool full = (base + CHUNK <= N);
  stage_chunk(t, band, yerr, y, sP, sA, sV, sY, amps, sigma2, invell, base, N);
  __syncthreads();
  if (full) thread_mobius<false>(sP, sA, sV, sigma2, base, N, mA);
  else      thread_mobius<true>(sP, sA, sV, sigma2, base, N, mA);
  __syncthreads();
  if (tid == 0) {
    float s = blockS0[blockIdx.x];
    for (int i = 0; i < BLOCK; ++i) { sS0[i] = s; s = mob_apply(mA + i * 4, s); }
  }
  __syncthreads();
  // Walk 1: per-thread linear matrices (needs sS0)
  {
    float l[4];
    if (full) walk_linear<false>(sP, sA, sV, sY, sigma2, sS0[tid], base, N, l);
    else      walk_linear<true>(sP, sA, sV, sY, sigma2, sS0[tid], base, N, l);
    mA[tid * 4 + 0] = l[0]; mA[tid * 4 + 1] = l[1];
    mA[tid * 4 + 2] = l[2]; mA[tid * 4 + 3] = l[3];
  }
  __syncthreads();
  if (tid == 0) {
    float g = blockG0[blockIdx.x];
    for (int i = 0; i < BLOCK; ++i) { sG0[i] = g; g = mob_apply(mA + i * 4, g); }
  }
  __syncthreads();
  // Walk 2: quad accumulation
  float quad;
  if (full) quad = walk_quad<false>(sP, sA, sV, sY, sigma2, sS0[tid], sG0[tid], base, N);
  else      quad = walk_quad<true>(sP, sA, sV, sY, sigma2, sS0[tid], sG0[tid], base, N);
  red[tid] = quad;
  __syncthreads();
  for (int s = BLOCK / 2; s > 0; s >>= 1) {
    if (tid < s) red[tid] += red[tid + s];
    __syncthreads();
  }
  if (tid == 0) blockQuad[blockIdx.x] = red[0];
}

// ---- Kernel F: final sum ---------------------------------------------------
__global__ void k_final(const float* __restrict__ blockLogD,
                        const float* __restrict__ blockQuad,
                        float* __restrict__ out, int nb, int N) {
  if (threadIdx.x == 0 && blockIdx.x == 0) {
    float ld = 0.0f, q = 0.0f;
    for (int k = 0; k < nb; ++k) { ld += blockLogD[k]; q += blockQuad[k]; }
    out[0] = -0.5f * (q + ld + (float)N * 1.8378770664093453f);  // N*log(2*pi)
  }
}

extern "C" void kernel_launch(void* const* d_in, const int* in_sizes, int n_in,
                              void* d_out, int out_size, void* d_ws, size_t ws_size,
                              hipStream_t stream) {
  const float* t    = (const float*)d_in[0];
  const int*   band = (const int*)d_in[1];
  const float* y    = (const float*)d_in[2];
  const float* yerr = (const float*)d_in[3];
  const float* lad  = (const float*)d_in[4];
  const float* lkp  = (const float*)d_in[5];
  float* out = (float*)d_out;
  int N = in_sizes[0];
  int nb = (N + CHUNK - 1) / CHUNK;   // 1024 for N = 2^21

  float* w = (float*)d_ws;            // ~48 KB of scratch
  float* blockMob  = w;               // nb*4
  float* blockS0   = blockMob + nb * 4;
  float* blockLin  = blockS0 + nb;    // nb*4
  float* blockG0   = blockLin + nb * 4;
  float* blockLogD = blockG0 + nb;
  float* blockQuad = blockLogD + nb;

  dim3 grid(nb), blk(BLOCK);
  k_blockA<<<grid, blk, 0, stream>>>(t, band, yerr, lad, lkp, blockMob, N);
  k_combS<<<1, 32, 0, stream>>>(blockMob, blockS0, nb);
  k_blockC<<<grid, blk, 0, stream>>>(t, band, y, yerr, lad, lkp, blockS0,
                                     blockLin, blockLogD, N);
  k_combG<<<1, 32, 0, stream>>>(blockLin, blockG0, nb);
  k_blockE<<<grid, blk, 0, stream>>>(t, band, y, yerr, lad, lkp, blockS0,
                                     blockG0, blockQuad, N);
  k_final<<<1, 32, 0, stream>>>(blockLogD, blockQuad, out, nb, N);
}